// StickBreaking_56762287784065
// MI455X (gfx1250) — compile-verified
//
#include <hip/hip_runtime.h>
#include <hip/hip_bf16.h>
#include <stdint.h>

// ---------------------------------------------------------------------------
// Stick-breaking on MI455X (gfx1250).
//
//  B = 1024 batches, N = 128.  Phase 1 (inverse SB + gate) is fully parallel;
//  phase 2 (forward SB) is a 16384-step serial scan per batch -> latency
//  bound, only the batch axis parallelizes.  No matmul structure exists
//  (clip/min/max scans), so WMMA does not apply; the CDNA5 paths used are the
//  async tensor/LDS data mover (global_load_async_to_lds_b128 /
//  global_store_async_from_lds_b128 + s_wait_asynccnt), wave32 shuffle scans,
//  and the 320KB LDS for per-lane column state.
// ---------------------------------------------------------------------------

#define NB    1024
#define NDIM  128
#define RS    132      // per-batch LDS row stride in floats (16B aligned, 2-way banked)

// ---- async global->LDS row loader: 32 x b128, immediate offsets -----------
// ISA: the immediate offset is added to both the LDS and the global address,
// so one immediate serves both sides (global chunk j == LDS chunk j).
template <int J>
__device__ __forceinline__ void async_row_load(uint32_t lds, uint64_t gaddr) {
  asm volatile("global_load_async_to_lds_b128 %0, %1, off offset:%2"
               :: "v"(lds), "v"(gaddr), "i"(J * 16) : "memory");
  if constexpr (J + 1 < NDIM / 4) async_row_load<J + 1>(lds, gaddr);
}

template <int J>
__device__ __forceinline__ void async_row_store(uint64_t gaddr, uint32_t lds) {
  asm volatile("global_store_async_from_lds_b128 %0, %1, off offset:%2"
               :: "v"(gaddr), "v"(lds), "i"(J * 16) : "memory");
  if constexpr (J + 1 < NDIM / 4) async_row_store<J + 1>(gaddr, lds);
}

__device__ __forceinline__ void wait_async0() {
  asm volatile("s_wait_asynccnt 0x0" ::: "memory");
}
__device__ __forceinline__ void wait_ds0() {
  asm volatile("s_wait_dscnt 0x0" ::: "memory");
}

// low 32 bits of a flat pointer to __shared__ == LDS offset on gfx1250
__device__ __forceinline__ uint32_t lds_off(const void* p) {
  return (uint32_t)(uintptr_t)p;
}

// ---------------------------------------------------------------------------
// Kernel 1: bgate = sigmoid(logits + logit(z)),  z = inv_stick_breaking(x).
// One wave per batch; lane owns columns 4*lane..4*lane+3.  All cumsums are
// running sums (rows) or 5-step wave32 shuffle scans (columns).
// ---------------------------------------------------------------------------
__global__ __launch_bounds__(32)
void sb_gate_kernel(const float* __restrict__ logits,
                    const float* __restrict__ mask,
                    float* __restrict__ bgate) {
  const int  lane = threadIdx.x;
  const size_t base = (size_t)blockIdx.x * (NDIM * NDIM) + (size_t)lane * 4;

  // pass A: column sums over rows (coalesced float4 row reads)
  float cs[4] = {0.f, 0.f, 0.f, 0.f};
  for (int m = 0; m < NDIM; ++m) {
    float4 mk = *reinterpret_cast<const float4*>(mask + base + (size_t)m * NDIM);
    cs[0] += mk.x; cs[1] += mk.y; cs[2] += mk.z; cs[3] += mk.w;
  }
  float icv[4];
#pragma unroll
  for (int j = 0; j < 4; ++j) icv[j] = 1.f / cs[j];

  float S[4] = {0.f, 0.f, 0.f, 0.f};   // exclusive row-cumsum of x per column
  for (int m = 0; m < NDIM; ++m) {
    const int mp = (m + 6 < NDIM) ? m + 6 : NDIM - 1;
    __builtin_prefetch(mask   + base + (size_t)mp * NDIM, 0, 0);
    __builtin_prefetch(logits + base + (size_t)mp * NDIM, 0, 0);

    float4 mk4 = *reinterpret_cast<const float4*>(mask   + base + (size_t)m * NDIM);
    float4 lg4 = *reinterpret_cast<const float4*>(logits + base + (size_t)m * NDIM);
    float mk[4] = {mk4.x, mk4.y, mk4.z, mk4.w};
    float lg[4] = {lg4.x, lg4.y, lg4.z, lg4.w};

    float x[4], fv[4];
#pragma unroll
    for (int j = 0; j < 4; ++j) {
      x[j]  = mk[j] * icv[j];
      fv[j] = fmaxf(0.f, mk[j] - S[j]);
    }
    // local prefixes over the lane's 4 columns
    const float px1 = x[0], px2 = px1 + x[1], px3 = px2 + x[2];
    const float xt  = px3 + x[3];
    const float f1 = fv[0], f2 = f1 + fv[1], f3 = f2 + fv[2], f4s = f3 + fv[3];

    // wave32 inclusive scans of lane totals
    float xi = xt, fi = f4s;
#pragma unroll
    for (int d = 1; d < 32; d <<= 1) {
      float a = __shfl_up(xi, d, 32);
      float c = __shfl_up(fi, d, 32);
      if (lane >= d) { xi += a; fi += c; }
    }
    const float xE = xi - xt;            // exclusive lane prefix (x)
    const float fE = fi - f4s;           // exclusive lane prefix (mfv)
    const float Ft = __shfl(fi, 31, 32); // row total of mfv

    const float pxl[4] = {0.f, px1, px2, px3};
    const float fil[4] = {f1, f2, f3, f4s};
    float og[4];
#pragma unroll
    for (int j = 0; j < 4; ++j) {
      const float rs  = xE + pxl[j];            // sum over cols < n
      const float mfm = Ft - (fE + fil[j]);     // sum over cols > n
      const float u1  = 1.f - rs;
      const float up  = mk[j] * fminf(u1, 1.f - S[j]);
      float       lo  = mk[j] * fmaxf(0.f, u1 - mfm);
      lo = fminf(lo, up);
      const float den = up - lo;
      float z = (den == 0.f) ? 0.f : (x[j] - lo) / den;   // divide_no_nan
      z = fminf(fmaxf(z, 1e-7f), 1.f - 1e-7f);
      // sigmoid(L + logit(z)) = z / (z + e^{-L} (1-z))
      og[j] = z / (z + __expf(-lg[j]) * (1.f - z));
      S[j] += x[j];
    }
    *reinterpret_cast<float4*>(bgate + base + (size_t)m * NDIM) =
        make_float4(og[0], og[1], og[2], og[3]);
  }
}

// ---------------------------------------------------------------------------
// Kernel 2: forward stick-breaking (serial).  Batch-per-lane, 32 batches per
// one-wave workgroup.  All hot data in LDS; rows double-buffered and moved
// with the async global<->LDS engine, stores pipelined one row behind so the
// single s_wait_asynccnt at the loop top hides both load and store latency.
// Reads bgate from `out` and overwrites it with p in place (row m of bgate
// is dead once row m of p is produced).
// ---------------------------------------------------------------------------
__global__ __launch_bounds__(32)
void sb_forward_kernel(const float* __restrict__ mask,
                       float* __restrict__ out) {
  __shared__ float sSp[32 * RS];        // per-batch column state S_p
  __shared__ float sMk[2][32 * RS];     // mask row, double buffered
  __shared__ float sBg[2][32 * RS];     // bgate row, double buffered
  __shared__ float sP [2][32 * RS];     // p row staging, double buffered

  const int lane = threadIdx.x;                       // lane == local batch
  const size_t NN   = (size_t)NDIM * NDIM;
  const size_t bofs = ((size_t)blockIdx.x * 32 + lane) * NN;
  const uint64_t gM = (uint64_t)(uintptr_t)(mask + bofs);
  const uint64_t gO = (uint64_t)(uintptr_t)(out  + bofs);

  const uint32_t lrow   = (uint32_t)(lane * RS * 4);
  const uint32_t lMk[2] = {lds_off(&sMk[0][0]) + lrow, lds_off(&sMk[1][0]) + lrow};
  const uint32_t lBg[2] = {lds_off(&sBg[0][0]) + lrow, lds_off(&sBg[1][0]) + lrow};
  const uint32_t lP [2] = {lds_off(&sP [0][0]) + lrow, lds_off(&sP [1][0]) + lrow};

  // prefetch row 0 while we zero the column state
  async_row_load<0>(lMk[0], gM);
  async_row_load<0>(lBg[0], gO);
  const float4 z4 = make_float4(0.f, 0.f, 0.f, 0.f);
  for (int n = 0; n < NDIM; n += 4)
    *reinterpret_cast<float4*>(&sSp[lane * RS + n]) = z4;

  for (int m = 0; m < NDIM; ++m) {
    const int cur = m & 1;
    wait_async0();                        // loads(m) + store(m-2) complete

    if (m > 0) {                          // issue store of previous row now,
      wait_ds0();                         // overlapped with this row's compute
      async_row_store<0>(gO + (uint64_t)(m - 1) * (NDIM * 4), lP[(m - 1) & 1]);
    }
    if (m + 1 < NDIM) {                   // prefetch next row
      const uint64_t off = (uint64_t)(m + 1) * (NDIM * 4);
      async_row_load<0>(lMk[cur ^ 1], gM + off);
      async_row_load<0>(lBg[cur ^ 1], gO + off);
    }

    const float* Mr = &sMk[cur][lane * RS];
    const float* Br = &sBg[cur][lane * RS];
    float*       Sp = &sSp[lane * RS];
    float*       Pr = &sP [cur][lane * RS];

    // pre-pass: T = sum_n max(0, mask - S_p)
    float T = 0.f;
    for (int n = 0; n < NDIM; n += 4) {
      float4 mk = *reinterpret_cast<const float4*>(&Mr[n]);
      float4 sp = *reinterpret_cast<const float4*>(&Sp[n]);
      T += fmaxf(0.f, mk.x - sp.x) + fmaxf(0.f, mk.y - sp.y)
         + fmaxf(0.f, mk.z - sp.z) + fmaxf(0.f, mk.w - sp.w);
    }

    // serial column scan (the true dependency chain)
    float r = 0.f, suf = T;
    for (int n = 0; n < NDIM; n += 4) {
      float4 mk4 = *reinterpret_cast<const float4*>(&Mr[n]);
      float4 sp4 = *reinterpret_cast<const float4*>(&Sp[n]);
      float4 bg4 = *reinterpret_cast<const float4*>(&Br[n]);
      const float mk[4] = {mk4.x, mk4.y, mk4.z, mk4.w};
      const float sp[4] = {sp4.x, sp4.y, sp4.z, sp4.w};
      const float bg[4] = {bg4.x, bg4.y, bg4.z, bg4.w};
      float pv[4], sv[4];
#pragma unroll
      for (int j = 0; j < 4; ++j) {
        const float fv = fmaxf(0.f, mk[j] - sp[j]);
        suf -= fv;                                   // mfm = sum over cols > n
        const float u1 = 1.f - r;
        const float up = mk[j] * fminf(u1, 1.f - sp[j]);
        float       lo = mk[j] * fmaxf(0.f, u1 - suf);
        lo = fminf(lo, up);
        float p = lo + bg[j] * (up - lo);
        p = fminf(fmaxf(p, 0.f), 1.f);
        r += p;
        pv[j] = p;
        sv[j] = sp[j] + p;
      }
      *reinterpret_cast<float4*>(&Pr[n]) = make_float4(pv[0], pv[1], pv[2], pv[3]);
      *reinterpret_cast<float4*>(&Sp[n]) = make_float4(sv[0], sv[1], sv[2], sv[3]);
    }
  }

  // flush the last row; S_ENDPGM performs an implicit wait-idle on all counters
  wait_ds0();
  async_row_store<0>(gO + (uint64_t)(NDIM - 1) * (NDIM * 4), lP[(NDIM - 1) & 1]);
}

extern "C" void kernel_launch(void* const* d_in, const int* in_sizes, int n_in,
                              void* d_out, int out_size, void* d_ws, size_t ws_size,
                              hipStream_t stream) {
  (void)in_sizes; (void)n_in; (void)out_size; (void)d_ws; (void)ws_size;
  const float* logits = (const float*)d_in[0];
  const float* mask   = (const float*)d_in[1];
  float*       out    = (float*)d_out;

  // Phase 1: parallel gate computation (bgate written into d_out).
  sb_gate_kernel<<<dim3(NB), dim3(32), 0, stream>>>(logits, mask, out);
  // Phase 2: serial forward stick-breaking, in place over d_out.
  sb_forward_kernel<<<dim3(NB / 32), dim3(32), 0, stream>>>(mask, out);
}